// MultiTaskGNN_7894149890557
// MI455X (gfx1250) — compile-verified
//
#include <hip/hip_runtime.h>
#include <hip/hip_bf16.h>

// ---------------------------------------------------------------------------
// MultiTaskGNN forward for MI455X (gfx1250, wave32, WMMA + TDM).
//   N=100000 nodes, E=600000 edges, H=128, 3 GraphSAGE layers + 4 MLP heads.
// Dense GEMMs on v_wmma_f32_16x16x4_f32; weight panels staged to LDS via the
// Tensor Data Mover (tensor_load_to_lds, TENSORcnt) when available.
// Workspace use: x[N*H] + agg[N*H] + deg[N] + invdeg[N]  ~= 103.2 MB.
// ---------------------------------------------------------------------------

#define NNODES 100000
#define NEDGES 600000
#define HDIM   128

typedef float v2f __attribute__((ext_vector_type(2)));
typedef float v8f __attribute__((ext_vector_type(8)));
typedef unsigned int u32x4 __attribute__((ext_vector_type(4)));
typedef int i32x4 __attribute__((ext_vector_type(4)));
typedef int i32x8 __attribute__((ext_vector_type(8)));

#if defined(__has_builtin)
#if __has_builtin(__builtin_amdgcn_tensor_load_to_lds) && \
    __has_builtin(__builtin_amdgcn_s_wait_tensorcnt)
#define USE_TDM 1
#endif
#endif

__device__ __forceinline__ v8f wmma4(v2f a, v2f b, v8f c) {
  // D(16x16,f32) = A(16x4,f32) x B(4x16,f32) + C
  return __builtin_amdgcn_wmma_f32_16x16x4_f32(false, a, false, b, (short)0, c,
                                               false, false);
}

#ifdef USE_TDM
// One TDM op: copy a [rows x 128] f32 panel (row stride 128) from global
// memory to LDS at byte offset lds_off.  D# layout per CDNA5 ISA ch.8.
__device__ __forceinline__ void tdm_load_panel(const float* g, unsigned lds_off,
                                               int rows) {
  unsigned long long ga = (unsigned long long)(size_t)g;
  u32x4 g0;
  g0.x = 1u;                                    // count=1 (valid), user mode
  g0.y = lds_off;                               // lds_addr (bytes)
  g0.z = (unsigned)(ga & 0xffffffffu);          // global_addr[31:0]
  g0.w = (unsigned)((ga >> 32) & 0x01ffffffu)   // global_addr[56:32]
         | (2u << 30);                          // type = 2 ("image")
  i32x8 g1;
  g1[0] = (int)(2u << 16);          // wg_mask=0, data_size=2 (4 bytes)
  g1[1] = (int)(128u << 16);        // tensor_dim0[15:0]=128 in bits[31:16]
  g1[2] = (int)((unsigned)rows << 16);  // td0 hi=0 | tensor_dim1[15:0]=rows
  g1[3] = (int)(128u << 16);        // td1 hi=0 | tile_dim0=128
  g1[4] = rows;                     // tile_dim1=rows, tile_dim2=0
  g1[5] = 128;                      // tensor_dim0_stride[31:0]=128
  g1[6] = 0;                        // stride0 hi | tensor_dim1_stride lo
  g1[7] = 0;                        // tensor_dim1_stride hi
  i32x4 z4 = {0, 0, 0, 0};
#if __clang_major__ >= 23
  i32x8 z8 = {0, 0, 0, 0, 0, 0, 0, 0};
  __builtin_amdgcn_tensor_load_to_lds(g0, g1, z4, z4, z8, 0);
#else
  __builtin_amdgcn_tensor_load_to_lds(g0, g1, z4, z4, 0);
#endif
}
#endif

// Stage [rows x 128] panels of two weight matrices into sB[0..8191] and
// sB[8192..16383] (float indices).
__device__ __forceinline__ void stage_panels(float* sB, const float* W0,
                                             const float* W1, int row0,
                                             int rows) {
#ifdef USE_TDM
  if ((threadIdx.x >> 5) == 0) {  // wave 0 issues both DMAs (TDM is per-wave)
    unsigned loff = (unsigned)(size_t)(void*)sB;  // flat low 32 = LDS offset
    tdm_load_panel(W0 + (size_t)row0 * HDIM, loff, rows);
    tdm_load_panel(W1 + (size_t)row0 * HDIM, loff + 8192u * 4u, rows);
  }
#else
#pragma unroll
  for (int t = 0; t < 16; ++t) {
    int ci = t * 256 + threadIdx.x;  // 0..4095 float4 chunks
    int mat = ci >> 11;
    int rem = ci & 2047;
    int r = rem >> 5;
    int c4 = (rem & 31) << 2;
    const float* src = (mat ? W1 : W0) + (size_t)(row0 + r) * HDIM + c4;
    *(float4*)&sB[mat * 8192 + r * HDIM + c4] = *(const float4*)src;
  }
  (void)rows;
#endif
}

__device__ __forceinline__ void stage_barrier() {
#ifdef USE_TDM
  __builtin_amdgcn_s_wait_tensorcnt(0);  // no-op for non-issuing waves
#endif
  __syncthreads();
}

// ---------------- node encoder: x = [static(6),p_obs,p_mask] @ W_enc + b ----
__global__ void k_encode(const float* __restrict__ node_static,
                         const float* __restrict__ p_obs,
                         const int*   __restrict__ p_mask,
                         const float* __restrict__ W_enc,
                         const float* __restrict__ b_enc,
                         float* __restrict__ x) {
  int i = blockIdx.x * blockDim.x + threadIdx.x;
  if (i >= NNODES * HDIM) return;
  int n = i >> 7, h = i & 127;
  float f[8];
#pragma unroll
  for (int j = 0; j < 6; ++j) f[j] = node_static[n * 6 + j];
  f[6] = p_obs[n];
  f[7] = (float)p_mask[n];
  float acc = b_enc[h];
#pragma unroll
  for (int j = 0; j < 8; ++j) acc += f[j] * W_enc[j * HDIM + h];
  x[i] = acc;
}

// ---------------- bidirectional degree -------------------------------------
__global__ void k_degree(const long long* __restrict__ ei,
                         float* __restrict__ deg) {
  int e = blockIdx.x * blockDim.x + threadIdx.x;
  if (e >= NEDGES) return;
  int s = (int)ei[e];
  int d = (int)ei[NEDGES + e];
  unsafeAtomicAdd(&deg[d], 1.0f);
  unsafeAtomicAdd(&deg[s], 1.0f);
}

__global__ void k_invdeg(const float* __restrict__ deg,
                         float* __restrict__ invd) {
  int n = blockIdx.x * blockDim.x + threadIdx.x;
  if (n < NNODES) invd[n] = 1.0f / fmaxf(deg[n], 1.0f);
}

// ---------------- mean-aggregation scatter (both directions) ---------------
__global__ void k_aggregate(const long long* __restrict__ ei,
                            const float* __restrict__ x,
                            float* __restrict__ agg) {
  unsigned gid = blockIdx.x * blockDim.x + threadIdx.x;  // < E*32
  int e = gid >> 5;
  if (e >= NEDGES) return;
  int c = (gid & 31) << 2;  // 4-float chunk
  int s = (int)ei[e];
  int d = (int)ei[NEDGES + e];
  float4 xs = *(const float4*)&x[(size_t)s * HDIM + c];
  float4 xd = *(const float4*)&x[(size_t)d * HDIM + c];
  float* ad = &agg[(size_t)d * HDIM + c];
  float* as = &agg[(size_t)s * HDIM + c];
  unsafeAtomicAdd(ad + 0, xs.x); unsafeAtomicAdd(ad + 1, xs.y);
  unsafeAtomicAdd(ad + 2, xs.z); unsafeAtomicAdd(ad + 3, xs.w);
  unsafeAtomicAdd(as + 0, xd.x); unsafeAtomicAdd(as + 1, xd.y);
  unsafeAtomicAdd(as + 2, xd.z); unsafeAtomicAdd(as + 3, xd.w);
}

// ---------------- SAGE layer: x = relu?(x@Ws + (agg*invd)@Wn + b) in-place --
// grid = N/16 blocks, 256 threads (8 waves); wave w owns output cols w*16..+15
__global__ void k_sage_layer(const float* __restrict__ Ws,
                             const float* __restrict__ Wn,
                             const float* __restrict__ bias,
                             const float* __restrict__ invd,
                             const float* __restrict__ agg,
                             float* __restrict__ x, int do_relu) {
  __shared__ float sB[2 * 64 * HDIM];  // 64 KB
  int tile = blockIdx.x;
  int wave = threadIdx.x >> 5;
  int lane = threadIdx.x & 31;
  int half = lane >> 4;
  int m    = lane & 15;
  int row  = tile * 16 + m;
  int col  = wave * 16 + m;
  int kb   = half * 2;

  const float* xrow = x   + (size_t)row * HDIM + kb;
  const float* arow = agg + (size_t)row * HDIM + kb;
  float iv = invd[row];

  v8f c = {};
  for (int ch = 0; ch < 2; ++ch) {
    __syncthreads();                       // previous panel fully consumed
    stage_panels(sB, Ws, Wn, ch * 64, 64);
    stage_barrier();
    const float* xc = xrow + ch * 64;
    const float* ac = arow + ch * 64;
#pragma unroll 4
    for (int k = 0; k < 64; k += 4) {
      float2 af = *(const float2*)(xc + k);
      v2f a = {af.x, af.y};
      v2f b = {sB[(k + kb) * HDIM + col], sB[(k + kb + 1) * HDIM + col]};
      c = wmma4(a, b, c);
      float2 ag = *(const float2*)(ac + k);
      v2f a2 = {ag.x * iv, ag.y * iv};
      v2f b2 = {sB[8192 + (k + kb) * HDIM + col],
                sB[8192 + (k + kb + 1) * HDIM + col]};
      c = wmma4(a2, b2, c);
    }
  }
  float bb = bias[col];
  __syncthreads();  // all reads of x in this block complete before overwrite
#pragma unroll
  for (int r = 0; r < 8; ++r) {
    float v = c[r] + bb;  // C frag: vgpr r, lane -> M = r + half*8, N = m
    if (do_relu) v = fmaxf(v, 0.0f);
    x[(size_t)(tile * 16 + half * 8 + r) * HDIM + col] = v;
  }
}

// ---------------- node heads: p_hat & node_logits --------------------------
__global__ void k_node_heads(const float* __restrict__ x,
                             const float* __restrict__ nrW1, const float* __restrict__ nrb1,
                             const float* __restrict__ nrW2, const float* __restrict__ nrb2,
                             const float* __restrict__ naW1, const float* __restrict__ nab1,
                             const float* __restrict__ naW2, const float* __restrict__ nab2,
                             float* __restrict__ p_hat,
                             float* __restrict__ node_logits) {
  __shared__ float sB[2 * 64 * HDIM];  // staging; reused as reduction scratch
  int tile = blockIdx.x;
  int wave = threadIdx.x >> 5;
  int lane = threadIdx.x & 31;
  int half = lane >> 4;
  int m    = lane & 15;
  int row  = tile * 16 + m;
  int col  = wave * 16 + m;
  int kb   = half * 2;

  const float* xrow = x + (size_t)row * HDIM + kb;
  v8f cnr = {}, cna = {};
  for (int ch = 0; ch < 2; ++ch) {
    __syncthreads();
    stage_panels(sB, nrW1, naW1, ch * 64, 64);
    stage_barrier();
    const float* xc = xrow + ch * 64;
#pragma unroll 4
    for (int k = 0; k < 64; k += 4) {
      float2 af = *(const float2*)(xc + k);
      v2f a = {af.x, af.y};
      v2f b1 = {sB[(k + kb) * HDIM + col], sB[(k + kb + 1) * HDIM + col]};
      cnr = wmma4(a, b1, cnr);
      v2f b2 = {sB[8192 + (k + kb) * HDIM + col],
                sB[8192 + (k + kb + 1) * HDIM + col]};
      cna = wmma4(a, b2, cna);
    }
  }
  float bnr = nrb1[col], bna = nab1[col];
  float wnr = nrW2[col], wna = naW2[col];

  __syncthreads();                       // done with staged weights
  if (threadIdx.x < 32) sB[threadIdx.x] = 0.0f;  // [0..15]=nr, [16..31]=na
  __syncthreads();
#pragma unroll
  for (int r = 0; r < 8; ++r) {
    float hnr = fmaxf(cnr[r] + bnr, 0.0f) * wnr;
    float hna = fmaxf(cna[r] + bna, 0.0f) * wna;
#pragma unroll
    for (int off = 1; off < 16; off <<= 1) {  // butterfly within 16-lane half
      hnr += __shfl_xor(hnr, off, 32);
      hna += __shfl_xor(hna, off, 32);
    }
    if (m == 0) {
      int lrow = half * 8 + r;
      unsafeAtomicAdd(&sB[lrow], hnr);
      unsafeAtomicAdd(&sB[16 + lrow], hna);
    }
  }
  __syncthreads();
  if (threadIdx.x < 16) {
    p_hat[tile * 16 + threadIdx.x]       = sB[threadIdx.x] + nrb2[0];
    node_logits[tile * 16 + threadIdx.x] = sB[16 + threadIdx.x] + nab2[0];
  }
}

// ---------------- edge heads: q_hat & edge_logits --------------------------
// edge feats = [x[src](128) | x[dst](128) | edge_static(4), q_obs, q_mask]
__global__ void k_edge_heads(const long long* __restrict__ ei,
                             const float* __restrict__ x,
                             const float* __restrict__ es,
                             const float* __restrict__ qo,
                             const int*   __restrict__ qm,
                             const float* __restrict__ erW1, const float* __restrict__ erb1,
                             const float* __restrict__ erW2, const float* __restrict__ erb2,
                             const float* __restrict__ eaW1, const float* __restrict__ eab1,
                             const float* __restrict__ eaW2, const float* __restrict__ eab2,
                             float* __restrict__ q_hat,
                             float* __restrict__ edge_logits) {
  __shared__ float sB[2 * 64 * HDIM];  // staging; reused as reduction scratch
  int tile = blockIdx.x;
  int wave = threadIdx.x >> 5;
  int lane = threadIdx.x & 31;
  int half = lane >> 4;
  int m    = lane & 15;
  int e    = tile * 16 + m;
  int col  = wave * 16 + m;
  int kb   = half * 2;

  int s = (int)ei[e];
  int d = (int)ei[NEDGES + e];
  const float* xs = x + (size_t)s * HDIM + kb;
  const float* xd = x + (size_t)d * HDIM + kb;

  v8f cer = {}, cea = {};
  // 4 chunks of 64 K-rows: {src 0-63, src 64-127, dst 0-63, dst 64-127}
  for (int ch = 0; ch < 4; ++ch) {
    __syncthreads();
    stage_panels(sB, erW1, eaW1, ch * 64, 64);
    stage_barrier();
    const float* ap = ((ch < 2) ? xs : xd) + (ch & 1) * 64;
#pragma unroll 4
    for (int k = 0; k < 64; k += 4) {
      float2 af = *(const float2*)(ap + k);
      v2f a = {af.x, af.y};
      v2f b1 = {sB[(k + kb) * HDIM + col], sB[(k + kb + 1) * HDIM + col]};
      cer = wmma4(a, b1, cer);
      v2f b2 = {sB[8192 + (k + kb) * HDIM + col],
                sB[8192 + (k + kb + 1) * HDIM + col]};
      cea = wmma4(a, b2, cea);
    }
  }
  // K rows 256..261 tail folded straight into the C fragment layout.
  float wtr[6], wta[6];
#pragma unroll
  for (int j = 0; j < 6; ++j) {
    wtr[j] = erW1[(256 + j) * HDIM + col];
    wta[j] = eaW1[(256 + j) * HDIM + col];
  }
#pragma unroll
  for (int r = 0; r < 8; ++r) {
    int er_ = tile * 16 + half * 8 + r;  // C frag row -> edge index
    float f0 = es[er_ * 4 + 0], f1 = es[er_ * 4 + 1];
    float f2 = es[er_ * 4 + 2], f3 = es[er_ * 4 + 3];
    float f4 = qo[er_], f5 = (float)qm[er_];
    cer[r] += f0 * wtr[0] + f1 * wtr[1] + f2 * wtr[2] +
              f3 * wtr[3] + f4 * wtr[4] + f5 * wtr[5];
    cea[r] += f0 * wta[0] + f1 * wta[1] + f2 * wta[2] +
              f3 * wta[3] + f4 * wta[4] + f5 * wta[5];
  }
  float ber = erb1[col], bea = eab1[col];
  float w2r = erW2[col], w2a = eaW2[col];

  __syncthreads();                       // done with staged weights
  if (threadIdx.x < 32) sB[threadIdx.x] = 0.0f;  // [0..15]=er, [16..31]=ea
  __syncthreads();
#pragma unroll
  for (int r = 0; r < 8; ++r) {
    float her = fmaxf(cer[r] + ber, 0.0f) * w2r;
    float hea = fmaxf(cea[r] + bea, 0.0f) * w2a;
#pragma unroll
    for (int off = 1; off < 16; off <<= 1) {
      her += __shfl_xor(her, off, 32);
      hea += __shfl_xor(hea, off, 32);
    }
    if (m == 0) {
      int lrow = half * 8 + r;
      unsafeAtomicAdd(&sB[lrow], her);
      unsafeAtomicAdd(&sB[16 + lrow], hea);
    }
  }
  __syncthreads();
  if (threadIdx.x < 16) {
    q_hat[tile * 16 + threadIdx.x]       = sB[threadIdx.x] + erb2[0];
    edge_logits[tile * 16 + threadIdx.x] = sB[16 + threadIdx.x] + eab2[0];
  }
}

// ---------------------------------------------------------------------------
extern "C" void kernel_launch(void* const* d_in, const int* in_sizes, int n_in,
                              void* d_out, int out_size, void* d_ws, size_t ws_size,
                              hipStream_t stream) {
  // setup_inputs() dict order:
  const long long* ei  = (const long long*)d_in[0];   // edge_index [2,E] int64
  const float* node_st = (const float*)d_in[1];       // [N,6]
  const float* edge_st = (const float*)d_in[2];       // [E,4]
  const float* p_obs   = (const float*)d_in[3];       // [N]
  const float* q_obs   = (const float*)d_in[4];       // [E]
  const int*   p_mask  = (const int*)d_in[5];         // [N] int32
  const int*   q_mask  = (const int*)d_in[6];         // [E] int32
  const float* W_enc   = (const float*)d_in[7];       // [8,128]
  const float* b_enc   = (const float*)d_in[8];       // [128]
  const float* W_self  = (const float*)d_in[9];       // [3,128,128]
  const float* W_neigh = (const float*)d_in[10];      // [3,128,128]
  const float* b_gnn   = (const float*)d_in[11];      // [3,128]
  const float* nrW1 = (const float*)d_in[12]; const float* nrb1 = (const float*)d_in[13];
  const float* nrW2 = (const float*)d_in[14]; const float* nrb2 = (const float*)d_in[15];
  const float* naW1 = (const float*)d_in[16]; const float* nab1 = (const float*)d_in[17];
  const float* naW2 = (const float*)d_in[18]; const float* nab2 = (const float*)d_in[19];
  const float* erW1 = (const float*)d_in[20]; const float* erb1 = (const float*)d_in[21];
  const float* erW2 = (const float*)d_in[22]; const float* erb2 = (const float*)d_in[23];
  const float* eaW1 = (const float*)d_in[24]; const float* eab1 = (const float*)d_in[25];
  const float* eaW2 = (const float*)d_in[26]; const float* eab2 = (const float*)d_in[27];

  // workspace carve-out (~103.2 MB)
  float* x    = (float*)d_ws;                  // [N*H]
  float* agg  = x   + (size_t)NNODES * HDIM;   // [N*H]
  float* deg  = agg + (size_t)NNODES * HDIM;   // [N]
  float* invd = deg + NNODES;                  // [N]

  // outputs: (p_hat[N], q_hat[E], node_logits[N], edge_logits[E]) flat
  float* p_hat       = (float*)d_out;
  float* q_hat       = p_hat + NNODES;
  float* node_logits = q_hat + NEDGES;
  float* edge_logits = node_logits + NNODES;

  (void)hipMemsetAsync(deg, 0, (size_t)NNODES * sizeof(float), stream);

  {
    int tot = NNODES * HDIM;
    k_encode<<<(tot + 255) / 256, 256, 0, stream>>>(node_st, p_obs, p_mask,
                                                    W_enc, b_enc, x);
  }
  k_degree<<<(NEDGES + 255) / 256, 256, 0, stream>>>(ei, deg);
  k_invdeg<<<(NNODES + 255) / 256, 256, 0, stream>>>(deg, invd);

  for (int l = 0; l < 3; ++l) {
    (void)hipMemsetAsync(agg, 0, (size_t)NNODES * HDIM * sizeof(float), stream);
    {
      unsigned tot = (unsigned)NEDGES * 32u;
      k_aggregate<<<(tot + 255) / 256, 256, 0, stream>>>(ei, x, agg);
    }
    k_sage_layer<<<NNODES / 16, 256, 0, stream>>>(
        W_self + (size_t)l * HDIM * HDIM, W_neigh + (size_t)l * HDIM * HDIM,
        b_gnn + (size_t)l * HDIM, invd, agg, x, (l < 2) ? 1 : 0);
  }

  k_node_heads<<<NNODES / 16, 256, 0, stream>>>(
      x, nrW1, nrb1, nrW2, nrb2, naW1, nab1, naW2, nab2, p_hat, node_logits);

  k_edge_heads<<<NEDGES / 16, 256, 0, stream>>>(
      ei, x, edge_st, q_obs, q_mask,
      erW1, erb1, erW2, erb2, eaW1, eab1, eaW2, eab2, q_hat, edge_logits);
}